// GraphSAGEBlock_73186242724262
// MI455X (gfx1250) — compile-verified
//
#include <hip/hip_runtime.h>
#include <hip/hip_bf16.h>

typedef float v2f __attribute__((ext_vector_type(2)));
typedef float v8f __attribute__((ext_vector_type(8)));

#define D 128              // D_IN == D_OUT == 128
#define TILES 8            // 128/16 output-feature tiles
#define WSTRIDE 132        // padded LDS row stride (dwords): 132 mod 64 = 4 -> conflict-free b64 reads
#define WMAT (D * WSTRIDE) // dwords per staged weight matrix (16896)

// ---------------------------------------------------------------------------
// Kernel 1: zero agg[N*128] and deg[N] (contiguous in workspace)
// ---------------------------------------------------------------------------
__global__ void sage_zero_kernel(float* __restrict__ p, int n) {
    int i = blockIdx.x * blockDim.x + threadIdx.x;
    int stride = gridDim.x * blockDim.x;
    for (; i < n; i += stride) p[i] = 0.0f;
}

// ---------------------------------------------------------------------------
// Kernel 2: edge scatter. One wave per edge: 32 lanes x float4 = 128 floats.
// Coalesced 512B gather of x[src], f32 global atomics into agg[dst] (L2 RMW).
// Edge indices pipelined one iteration ahead; next x-row prefetched
// (global_prefetch_b8) to hide the dependent gather latency.
// ---------------------------------------------------------------------------
__global__ __launch_bounds__(256) void sage_scatter_kernel(
    const float* __restrict__ x, const int* __restrict__ ei,
    float* __restrict__ agg, float* __restrict__ deg, int E) {
    const int lane = threadIdx.x & 31;
    const int warp0 = (blockIdx.x * blockDim.x + threadIdx.x) >> 5;
    const int nwarps = (gridDim.x * blockDim.x) >> 5;

    int e = warp0;
    if (e >= E) return;
    int src = ei[e];
    int dst = ei[E + e];

    while (e < E) {
        const int en = e + nwarps;
        int srcn = 0, dstn = 0;
        if (en < E) {
            srcn = ei[en];
            dstn = ei[E + en];
            __builtin_prefetch(x + (size_t)srcn * D + lane * 4, 0, 3);
        }
        const float4* xs = (const float4*)(x + (size_t)src * D);
        float4 v = xs[lane];
        float* ad = agg + (size_t)dst * D + lane * 4;
        atomicAdd(ad + 0, v.x);
        atomicAdd(ad + 1, v.y);
        atomicAdd(ad + 2, v.z);
        atomicAdd(ad + 3, v.w);
        if (lane == 0) atomicAdd(deg + dst, 1.0f);
        e = en; src = srcn; dst = dstn;
    }
}

// ---------------------------------------------------------------------------
// Kernel 3: fused  out = LN( (agg/deg)@Wl^T + bl + x@Wr^T ) -> ReLU
// Weights staged once per block into LDS (132KB, padded stride 132 dwords);
// all 8 waves serve B fragments via conflict-free ds_load_b64.
// One wave = 16 nodes x 128 out-features via v_wmma_f32_16x16x4_f32.
// A-row index is CLAMPED (not predicated) so EXEC stays all-1s and A loads
// issue unconditionally; out-of-range rows never reach the guarded stores.
// ---------------------------------------------------------------------------
__global__ __launch_bounds__(256) void sage_wmma_kernel(
    const float* __restrict__ agg, const float* __restrict__ deg,
    const float* __restrict__ x,
    const float* __restrict__ Wl, const float* __restrict__ bl,
    const float* __restrict__ Wr,
    const float* __restrict__ gamma, const float* __restrict__ beta,
    float* __restrict__ out, int N) {

    extern __shared__ float ldsW[];    // [2][128][WSTRIDE]

    const int lane = threadIdx.x & 31;
    const int wave = threadIdx.x >> 5;
    const int rowBase = (blockIdx.x * 8 + wave) * 16;

    // ---- stage Wl, Wr into LDS with padded rows (all 256 threads) ----
    for (int idx = threadIdx.x; idx < D * D; idx += 256) {
        const int r = idx >> 7, c = idx & (D - 1);
        ldsW[r * WSTRIDE + c]        = Wl[idx];
        ldsW[WMAT + r * WSTRIDE + c] = Wr[idx];
    }
    __syncthreads();

    if (rowBase >= N) return;          // wave-uniform: EXEC all-1s for WMMA

    const int half = lane >> 4;        // 0: lanes 0-15, 1: lanes 16-31
    const int l16  = lane & 15;

    // A-fragment source row (clamped; garbage rows are store-guarded later)
    int arow = rowBase + l16;
    arow = (arow < N) ? arow : (N - 1);
    const float invdeg = 1.0f / fmaxf(deg[arow], 1.0f);
    const float* aggRow = agg + (size_t)arow * D + 2 * half;
    const float* xRow   = x   + (size_t)arow * D + 2 * half;

    // per-lane base pointers into staged weights: row o = 16t + l16
    const float* lwlBase = ldsW        + (size_t)l16 * WSTRIDE + 2 * half;
    const float* lwrBase = ldsW + WMAT + (size_t)l16 * WSTRIDE + 2 * half;

    v8f acc[TILES];
    for (int t = 0; t < TILES; ++t)
        for (int j = 0; j < 8; ++j) acc[t][j] = 0.0f;

    for (int k0 = 0; k0 < D; k0 += 4) {
        // streaming, read-once -> non-temporal loads (TH=NT)
        v2f rawA = __builtin_nontemporal_load((const v2f*)(aggRow + k0));
        v2f rawX = __builtin_nontemporal_load((const v2f*)(xRow + k0));
        v2f aA;  aA[0] = rawA[0] * invdeg;  aA[1] = rawA[1] * invdeg;

        for (int t = 0; t < TILES; ++t) {
            // B[k][o] = W[o][k], rows 16t+l16, cols k0+2h..+1 (conflict-free)
            const v2f bL = *(const v2f*)(lwlBase + (size_t)t * 16 * WSTRIDE + k0);
            const v2f bR = *(const v2f*)(lwrBase + (size_t)t * 16 * WSTRIDE + k0);
            acc[t] = __builtin_amdgcn_wmma_f32_16x16x4_f32(
                false, aA, false, bL, (short)0, acc[t], false, false);
            acc[t] = __builtin_amdgcn_wmma_f32_16x16x4_f32(
                false, rawX, false, bR, (short)0, acc[t], false, false);
        }
    }

    // epilogue params per lane/tile
    float bl_o[TILES], g_o[TILES], be_o[TILES];
    for (int t = 0; t < TILES; ++t) {
        const int o = t * 16 + l16;
        bl_o[t] = bl[o]; g_o[t] = gamma[o]; be_o[t] = beta[o];
    }

    // bias add + LayerNorm statistics (row m spread over 16 lanes x 8 tiles)
    float s1[8], s2[8];
    for (int r = 0; r < 8; ++r) { s1[r] = 0.0f; s2[r] = 0.0f; }
    for (int t = 0; t < TILES; ++t)
        for (int r = 0; r < 8; ++r) {
            float v = acc[t][r] + bl_o[t];
            acc[t][r] = v;
            s1[r] += v;  s2[r] += v * v;
        }
    for (int off = 8; off >= 1; off >>= 1)
        for (int r = 0; r < 8; ++r) {
            s1[r] += __shfl_xor(s1[r], off, 16);   // width-16 keeps halves apart
            s2[r] += __shfl_xor(s2[r], off, 16);
        }
    float mu[8], rs[8];
    for (int r = 0; r < 8; ++r) {
        mu[r] = s1[r] * (1.0f / 128.0f);
        float var = s2[r] * (1.0f / 128.0f) - mu[r] * mu[r];
        rs[r] = rsqrtf(var + 1e-5f);
    }

    // normalize + affine + ReLU + store (lanes 0-15: rows r, 16-31: rows r+8)
    for (int r = 0; r < 8; ++r) {
        const int row = rowBase + r + 8 * half;
        if (row >= N) continue;
        float* orow = out + (size_t)row * D;
        for (int t = 0; t < TILES; ++t) {
            float y = (acc[t][r] - mu[r]) * rs[r] * g_o[t] + be_o[t];
            y = fmaxf(y, 0.0f);
            __builtin_nontemporal_store(y, orow + t * 16 + l16);  // write-once
        }
    }
}

// ---------------------------------------------------------------------------
extern "C" void kernel_launch(void* const* d_in, const int* in_sizes, int n_in,
                              void* d_out, int out_size, void* d_ws, size_t ws_size,
                              hipStream_t stream) {
    const float* x     = (const float*)d_in[0];
    const int*   ei    = (const int*)d_in[1];
    const float* Wl    = (const float*)d_in[2];
    const float* blv   = (const float*)d_in[3];
    const float* Wr    = (const float*)d_in[4];
    const float* gamma = (const float*)d_in[5];
    const float* beta  = (const float*)d_in[6];
    float*       out   = (float*)d_out;

    const int N = in_sizes[0] / D;     // 100000
    const int E = in_sizes[1] / 2;     // 1600000

    float* agg = (float*)d_ws;         // [N,128]
    float* deg = agg + (size_t)N * D;  // [N]  (contiguous -> one zero pass)

    const int zn = N * (D + 1);
    sage_zero_kernel<<<(zn + 255) / 256, 256, 0, stream>>>(agg, zn);

    sage_scatter_kernel<<<1024, 256, 0, stream>>>(x, ei, agg, deg, E);

    const int    blocks   = (N + 127) / 128;            // 8 waves/block, 16 rows/wave
    const size_t ldsBytes = 2u * WMAT * sizeof(float);  // 132 KB staged weights
    sage_wmma_kernel<<<blocks, 256, ldsBytes, stream>>>(
        agg, deg, x, Wl, blv, Wr, gamma, beta, out, N);
}